// PhiFromAttention_84902913507650
// MI455X (gfx1250) — compile-verified
//
#include <hip/hip_runtime.h>
#include <hip/hip_bf16.h>
#include <math.h>

#define EPSV 1e-6f

typedef __attribute__((ext_vector_type(2))) float v2f;
typedef __attribute__((ext_vector_type(8))) float v8f;

// LDS "extras" layout (floats):
#define EX_LOGDET  0   // logdet accumulator (f32)
#define EX_NEG     1   // (int) count of negative pivots
#define EX_ZERO    2   // (int) zero-pivot flag
#define EX_MI      3   // broadcast slot for resulting MI
#define EX_LOGDIAG 4   // sum log(|diag(cov)|+eps) accumulator
#define EX_COUNT   8

// Byte offset of the dynamic-LDS region (starts after static LDS).
__device__ __forceinline__ unsigned dyn_lds_base() {
  return __builtin_amdgcn_groupstaticsize();
}

// Async copy: 16 bytes global -> LDS, tracked by ASYNCcnt (no VGPR staging).
__device__ __forceinline__ void async_g2l_b128(unsigned lds_off, const void* gaddr) {
  asm volatile("global_load_async_to_lds_b128 %0, %1, off"
               :: "v"(lds_off), "v"((unsigned long long)(uintptr_t)gaddr)
               : "memory");
}
// Async copy: 4 bytes global -> LDS (per-lane gather into LDS).
__device__ __forceinline__ void async_g2l_b32(unsigned lds_off, const void* gaddr) {
  asm volatile("global_load_async_to_lds_b32 %0, %1, off"
               :: "v"(lds_off), "v"((unsigned long long)(uintptr_t)gaddr)
               : "memory");
}
__device__ __forceinline__ void wait_async0() {
  asm volatile("s_wait_asynccnt 0x0" ::: "memory");
}

// ---------------------------------------------------------------------------
// Blocked LDL^T factorization of an NMAT x NMAT symmetric matrix held in LDS,
// returning the clamped Gaussian-MI value 0.5*(logdiag - logdet) (slogdet
// semantics: result forced to 0 when det sign <= 0).  Trailing Schur updates
// run on v_wmma_f32_16x16x4_f32.  Must be called by all 256 threads.
// ---------------------------------------------------------------------------
template<int NMAT>
__device__ float ldl_mi(float* __restrict__ S, float* __restrict__ ex) {
  const int tid   = threadIdx.x;
  constexpr int T = NMAT / 16;
  const int lane  = tid & 31;
  // readfirstlane: provably wave-uniform wave id -> scalar control flow around
  // the WMMA region (EXEC stays all-ones, as the ISA requires).
  const int wave  = __builtin_amdgcn_readfirstlane((int)(threadIdx.x >> 5));
  const int mrow  = lane & 15;   // M (or N) index of this lane's fragment slice
  const int khalf = lane >> 4;   // 0: lanes 0-15, 1: lanes 16-31

  for (int k = 0; k < T; ++k) {
    float* Dk = S + (k * 16) * NMAT + k * 16;

    // (1) serial LDL^T of the 16x16 diagonal tile; accumulate slogdet state
    if (tid == 0) {
      float logacc = 0.f; int neg = 0; int zf = 0;
      for (int c = 0; c < 16; ++c) {
        float d = Dk[c * NMAT + c];
        for (int t = 0; t < c; ++t) {
          const float l = Dk[c * NMAT + t];
          d -= l * l * Dk[t * NMAT + t];
        }
        if (d < 0.f) neg++;
        float ad = fabsf(d);
        if (ad < 1e-30f) { zf = 1; d = (d < 0.f) ? -1e-30f : 1e-30f; ad = 1e-30f; }
        logacc += __logf(ad);
        Dk[c * NMAT + c] = d;                       // store pivot d_c
        for (int r = c + 1; r < 16; ++r) {          // unit-lower L in place
          float v = Dk[r * NMAT + c];
          for (int t = 0; t < c; ++t)
            v -= Dk[r * NMAT + t] * Dk[t * NMAT + t] * Dk[c * NMAT + t];
          Dk[r * NMAT + c] = v / d;
        }
      }
      ex[EX_LOGDET] += logacc;
      ((int*)ex)[EX_NEG]  += neg;
      ((int*)ex)[EX_ZERO] |= zf;
    }
    __syncthreads();

    const int rem = T - 1 - k;
    if (rem > 0) {
      // (2) panel triangular solve: L_ik = A_ik * L_kk^-T * D^-1 (row per thread)
      const int nrows = rem * 16;
      for (int r = tid; r < nrows; r += 256) {
        float* Pr = S + ((k + 1) * 16 + r) * NMAT + k * 16;
        #pragma unroll 4
        for (int c = 0; c < 16; ++c) {
          float v = Pr[c];
          for (int t = 0; t < c; ++t)
            v -= Pr[t] * Dk[t * NMAT + t] * Dk[c * NMAT + t];
          Pr[c] = v / Dk[c * NMAT + c];
        }
      }
      __syncthreads();

      // (3) trailing update A_ij -= L_ik * D * L_jk^T   (WMMA f32 16x16x4)
      // Tile (ii,jj), jj<=ii, one 16x16 tile per wave iteration.
      const int ntiles = rem * (rem + 1) / 2;
      for (int t = wave; t < ntiles; t += 8) {
        int tt = t, ii = 0;
        while (tt >= ii + 1) { tt -= ii + 1; ii++; }
        const int gi = k + 1 + ii;
        const int gj = k + 1 + tt;
        float*       Cb = S + (gi * 16) * NMAT + gj * 16;
        const float* Ar = S + (gi * 16 + mrow) * NMAT + k * 16;  // row of L_ik
        const float* Br = S + (gj * 16 + mrow) * NMAT + k * 16;  // row of L_jk

        // C fragment: VGPR r -> M = r + khalf*8, N = mrow
        v8f c;
        #pragma unroll
        for (int r = 0; r < 8; ++r) c[r] = Cb[(r + khalf * 8) * NMAT + mrow];

        #pragma unroll
        for (int kk = 0; kk < 16; kk += 4) {
          const int K0 = kk + khalf * 2;
          v2f a, b;
          // A fragment (16x4): M = mrow, K = K0/K0+1; A = -L_ik * D
          a[0] = -Ar[K0]     * Dk[K0 * NMAT + K0];
          a[1] = -Ar[K0 + 1] * Dk[(K0 + 1) * NMAT + (K0 + 1)];
          // B fragment (4x16): N = mrow, K = K0/K0+1; B[K][N] = L_jk[N][K]
          b[0] = Br[K0];
          b[1] = Br[K0 + 1];
          c = __builtin_amdgcn_wmma_f32_16x16x4_f32(
              /*neg_a=*/false, a, /*neg_b=*/false, b,
              /*c_mod=*/(short)0, c, /*reuse_a=*/false, /*reuse_b=*/false);
        }
        #pragma unroll
        for (int r = 0; r < 8; ++r) Cb[(r + khalf * 8) * NMAT + mrow] = c[r];
      }
      __syncthreads();
    }
  }

  if (tid == 0) {
    const float mi  = 0.5f * (ex[EX_LOGDIAG] - ex[EX_LOGDET]);
    const bool  pos = ((((int*)ex)[EX_NEG] & 1) == 0) && (((int*)ex)[EX_ZERO] == 0);
    ex[EX_MI] = pos ? fmaxf(mi, 0.f) : 0.f;
  }
  __syncthreads();
  return ex[EX_MI];
}

// ---------------------------------------------------------------------------
// Kernel 1: full 256x256 MI per matrix.  cov (256 KB) lives entirely in the
// WGP's 320 KB LDS.  One workgroup per matrix.  The row-major term of
// cov = m + m^T is DMA'd straight into LDS with async global->LDS B128 ops.
// ---------------------------------------------------------------------------
__global__ __launch_bounds__(256) void phi_full_kernel(
    const float* __restrict__ attn, float* __restrict__ Itot) {
  extern __shared__ float smem[];
  float* S  = smem;               // 256*256 floats
  float* ex = smem + 256 * 256;   // EX_COUNT floats
  const int g   = blockIdx.x;
  const int tid = threadIdx.x;
  const float* M = attn + (size_t)g * 256 * 256;

  if (tid < EX_COUNT) ex[tid] = 0.f;

  // Stage 1: S[e] = m[e] via async copy (16B per op, 64 ops/thread, no VGPRs).
  {
    const unsigned base = dyn_lds_base();
    for (int e = tid * 4; e < 256 * 256; e += 256 * 4)
      async_g2l_b128(base + (unsigned)e * 4u, M + e);
    wait_async0();
  }
  __syncthreads();

  // Stage 2: symmetrize (+eps on diag, accumulate log|diag|).
  for (int e = tid; e < 256 * 256; e += 256) {
    const int i = e >> 8, j = e & 255;
    float cv = S[e] + M[j * 256 + i];
    if (i == j) {
      atomicAdd(&ex[EX_LOGDIAG], __logf(fabsf(cv) + EPSV));
      cv += EPSV;               // eps added to cov AFTER diag extraction
    }
    S[e] = cv;
  }
  __syncthreads();

  const float mi = ldl_mi<256>(S, ex);
  if (tid == 0) Itot[g] = mi;
}

// ---------------------------------------------------------------------------
// Kernel 2: per (matrix, permutation) partition MI.  Gathers both 128x128
// permuted sub-covariances from L2-resident global memory: the first term is
// scattered straight into LDS with per-lane async B32 loads, the transposed
// term is added in a second pass.  Atomic-min of (mi_a + mi_b) per matrix.
// ---------------------------------------------------------------------------
__global__ __launch_bounds__(256) void phi_perm_kernel(
    const float* __restrict__ attn, const int* __restrict__ perms,
    unsigned* __restrict__ minbits) {
  extern __shared__ float smem[];
  float* S   = smem;                                   // 128*128 floats
  float* ex  = smem + 128 * 128;                       // EX_COUNT floats
  int*   idx = (int*)(smem + 128 * 128 + EX_COUNT);    // 128 ints
  const int g   = blockIdx.x;
  const int p   = blockIdx.y;
  const int tid = threadIdx.x;
  const float* M  = attn + (size_t)g * 256 * 256;
  const int*   pg = perms + ((size_t)g * 10 + p) * 256;

  float misum = 0.f;
  for (int half = 0; half < 2; ++half) {
    if (tid < EX_COUNT) ex[tid] = 0.f;
    if (tid < 128) {
      const int r = pg[half * 128 + tid];
      idx[tid] = r;
      __builtin_prefetch(&M[(size_t)r * 256], 0, 1);   // global_prefetch_b8
    }
    __syncthreads();

    // Pass 1: S[e] = m[pa_i, pa_j]  (async per-lane gather into LDS).
    {
      const unsigned base = dyn_lds_base();
      for (int e = tid; e < 128 * 128; e += 256) {
        const int i = e >> 7, j = e & 127;
        async_g2l_b32(base + (unsigned)e * 4u, M + idx[i] * 256 + idx[j]);
      }
      wait_async0();
    }
    __syncthreads();

    // Pass 2: add transposed term, eps on diag, accumulate log|diag|.
    for (int e = tid; e < 128 * 128; e += 256) {
      const int i = e >> 7, j = e & 127;
      float cv = S[e] + M[idx[j] * 256 + idx[i]];
      if (i == j) {
        atomicAdd(&ex[EX_LOGDIAG], __logf(fabsf(cv) + EPSV));
        cv += EPSV;
      }
      S[e] = cv;
    }
    __syncthreads();

    misum += ldl_mi<128>(S, ex);
    __syncthreads();   // protect ex/idx reuse in next half
  }

  if (tid == 0) atomicMin(&minbits[g], __float_as_uint(misum));  // misum >= 0
}

// ---------------------------------------------------------------------------
// Init + finalize
// ---------------------------------------------------------------------------
__global__ void init_kernel(unsigned* __restrict__ minbits, int n) {
  const int i = blockIdx.x * blockDim.x + threadIdx.x;
  if (i < n) minbits[i] = 0x7f7fffffu;   // FLT_MAX bits
}

__global__ __launch_bounds__(512) void finalize_kernel(
    const float* __restrict__ Itot, const unsigned* __restrict__ minbits,
    float* __restrict__ out) {
  __shared__ float phi[512];
  const int t = threadIdx.x;             // t indexes (b*16 + h)
  const float mn = __uint_as_float(minbits[t]);
  const float ph = fmaxf(0.f, Itot[t] - mn);
  phi[t] = tanhf(ph * (1.0f / 25.6f));   // max(1, 256*0.1) = 25.6
  __syncthreads();
  if (t < 32) {
    float s = 0.f;
    #pragma unroll
    for (int h = 0; h < 16; ++h) s += phi[t * 16 + h];
    out[t] = s * (1.0f / 16.0f);
  }
}

// ---------------------------------------------------------------------------
extern "C" void kernel_launch(void* const* d_in, const int* in_sizes, int n_in,
                              void* d_out, int out_size, void* d_ws, size_t ws_size,
                              hipStream_t stream) {
  (void)in_sizes; (void)n_in; (void)out_size; (void)ws_size;
  const float* attn  = (const float*)d_in[0];   // [32,16,256,256] f32
  const int*   perms = (const int*)d_in[1];     // [512,10,256] i32
  float*    out     = (float*)d_out;            // [32] f32
  float*    Itot    = (float*)d_ws;             // 512 floats
  unsigned* minbits = (unsigned*)((char*)d_ws + 512 * sizeof(float));

  const int FULL_LDS = (256 * 256 + EX_COUNT) * (int)sizeof(float);           // 262176 B
  const int PERM_LDS = (128 * 128 + EX_COUNT) * (int)sizeof(float) + 128 * 4; // 66080 B

  // Opt in to >64KB dynamic LDS (CDNA5 WGP has 320KB).
  hipFuncSetAttribute(reinterpret_cast<const void*>(phi_full_kernel),
                      hipFuncAttributeMaxDynamicSharedMemorySize, FULL_LDS);
  hipFuncSetAttribute(reinterpret_cast<const void*>(phi_perm_kernel),
                      hipFuncAttributeMaxDynamicSharedMemorySize, PERM_LDS);

  init_kernel<<<2, 256, 0, stream>>>(minbits, 512);
  phi_full_kernel<<<512, 256, FULL_LDS, stream>>>(attn, Itot);
  phi_perm_kernel<<<dim3(512, 10), 256, PERM_LDS, stream>>>(attn, perms, minbits);
  finalize_kernel<<<1, 512, 0, stream>>>(Itot, minbits, out);
}